// VFELayer_13331578487174
// MI455X (gfx1250) — compile-verified
//
#include <hip/hip_runtime.h>
#include <math.h>

// Problem constants (match reference)
#define KVOX   100000
#define TPTS   35
#define CIN    7
#define CAGG   16
#define NROWS  (KVOX * TPTS)      // 3,500,000 rows of the flattened [K*T, CIN] matrix
#define NTILES (NROWS / 16)       // 218,750 — divides exactly, no row tail
#define BN_EPS 1e-5f

typedef float v2f __attribute__((ext_vector_type(2)));
typedef float v8f __attribute__((ext_vector_type(8)));

// ws layout (floats): [0..15] sum, [16..31] sumsq, [32..47] scale, [48..63] shift
__global__ void vfe_zero_stats(float* __restrict__ stats) {
    if (threadIdx.x < 32) stats[threadIdx.x] = 0.0f;
}

// ---------------------------------------------------------------------------
// Pass 1: per-channel sum / sumsq of relu(inputs @ W^T + b) over all 3.5M rows.
// GEMM via V_WMMA_F32_16X16X4_F32: two chained WMMAs cover K=7 (zero-padded 8).
// A (16x4 f32) lane layout: lanes 0-15 -> M=lane, K={0,1}; lanes 16-31 -> K={2,3}.
// B (4x16 f32): N = lane%16, same K split. C(M,N): a lane's 8 C-VGPRs all share
// one column N=lane%16 -> per-lane scalar channel accumulators.
// All pad-lane values produced with unconditional loads + selects (no EXEC
// divergence in the hot loop; WMMA requires EXEC all-ones).
// ---------------------------------------------------------------------------
__global__ void vfe_stats(const float* __restrict__ in,   // [NROWS, 7]
                          const float* __restrict__ W,    // [16, 7]
                          const float* __restrict__ bias, // [16]
                          float* __restrict__ stats) {
    __shared__ float smem[32];
    const int lane  = threadIdx.x & 31;
    const int ch    = lane & 15;       // N (channel) for B/C; M row-in-tile for A
    const int khalf = lane >> 4;       // 0: K={0,1}, 1: K={2,3}
    const int koff  = 2 * khalf;       // 0 or 2

    if (threadIdx.x < 32) smem[threadIdx.x] = 0.0f;
    __syncthreads();

    // B-matrix registers: B[k][n] = W[n][k] (einsum ktc,dc). Constant over loop.
    const int wb = ch * CIN;
    v2f bm1, bm2;
    bm1.x = W[wb + koff];                  // K = 0 or 2
    bm1.y = W[wb + koff + 1];              // K = 1 or 3
    bm2.x = W[wb + 4 + koff];              // K = 4 or 6
    {
        const float w5 = W[wb + 5];        // valid for all lanes
        bm2.y = (khalf == 0) ? w5 : 0.0f;  // K = 5 or 7 (zero pad), select not branch
    }

    const float bseed = bias[ch];          // seed C accumulator with per-channel bias

    const int wave   = blockIdx.x * (blockDim.x >> 5) + (threadIdx.x >> 5);
    const int nwaves = gridDim.x * (blockDim.x >> 5);

    float sum = 0.0f, sumsq = 0.0f;

    for (int tile = wave; tile < NTILES; tile += nwaves) {  // wave-uniform trip count
        const int  row  = tile * 16 + ch;      // M = ch for A
        const long base = (long)row * CIN;
        v2f a1, a2;
        a1.x = in[base + koff];
        a1.y = in[base + koff + 1];
        a2.x = in[base + 4 + koff];            // khalf=1 -> base+6 (valid)
        const float t5 = in[base + 5];         // valid for all lanes
        a2.y = (khalf == 0) ? t5 : 0.0f;       // zero pad via cndmask

        v8f c;
#pragma unroll
        for (int j = 0; j < 8; ++j) c[j] = bseed;
        c = __builtin_amdgcn_wmma_f32_16x16x4_f32(false, a1, false, bm1,
                                                  (short)0, c, false, false);
        c = __builtin_amdgcn_wmma_f32_16x16x4_f32(false, a2, false, bm2,
                                                  (short)0, c, false, false);
#pragma unroll
        for (int j = 0; j < 8; ++j) {
            float x = fmaxf(c[j], 0.0f);       // relu
            sum   += x;
            sumsq  = fmaf(x, x, sumsq);
        }
    }

    // Block reduction in LDS (ds_add_f32), then 32 global atomics per block.
    atomicAdd(&smem[ch], sum);
    atomicAdd(&smem[16 + ch], sumsq);
    __syncthreads();
    if (threadIdx.x < 32) atomicAdd(&stats[threadIdx.x], smem[threadIdx.x]);
}

// Pass 1.5: fold BN into xn = x*scale + shift (biased variance, matches jnp).
__global__ void vfe_finalize(const float* __restrict__ gamma,
                             const float* __restrict__ beta,
                             float* __restrict__ stats) {
    const int d = threadIdx.x;
    if (d < 16) {
        const float inv_n = 1.0f / (float)NROWS;
        const float mean  = stats[d] * inv_n;
        const float var   = stats[16 + d] * inv_n - mean * mean;
        const float scale = gamma[d] / sqrtf(var + BN_EPS);
        stats[32 + d] = scale;
        stats[48 + d] = beta[d] - mean * scale;
    }
}

// ---------------------------------------------------------------------------
// Pass 2: recompute x (inputs mostly L2-resident after pass 1; 192 MB L2),
// normalize, max over T, write [K,T,32] masked output. Half-wave (16 lanes) =
// one voxel, lane = channel -> 16 consecutive f32 stores = 64B coalesced.
// This pass is the bandwidth floor: 448 MB of stores at 23.3 TB/s.
// ---------------------------------------------------------------------------
__global__ void vfe_output(const float* __restrict__ in,    // [K,T,7]
                           const int*   __restrict__ mask,  // [K,T]
                           const float* __restrict__ W,     // [16,7]
                           const float* __restrict__ bias,  // [16]
                           const float* __restrict__ stats,
                           float* __restrict__ out) {       // [K,T,32]
    const int lane = threadIdx.x & 31;
    const int d    = lane & 15;
    const int half = lane >> 4;
    const int wave = blockIdx.x * (blockDim.x >> 5) + (threadIdx.x >> 5);
    const int k    = wave * 2 + half;
    if (k >= KVOX) return;

    const float w0 = W[d * CIN + 0], w1 = W[d * CIN + 1], w2 = W[d * CIN + 2];
    const float w3 = W[d * CIN + 3], w4 = W[d * CIN + 4], w5 = W[d * CIN + 5];
    const float w6 = W[d * CIN + 6];
    const float bv    = bias[d];
    const float scale = stats[32 + d];
    const float shift = stats[48 + d];

    float agg = -INFINITY;
#pragma unroll 5
    for (int t = 0; t < TPTS; ++t) {
        const long r = (long)k * TPTS + t;
        const float* p = in + r * CIN;
        float x = bv;
        x = fmaf(p[0], w0, x); x = fmaf(p[1], w1, x); x = fmaf(p[2], w2, x);
        x = fmaf(p[3], w3, x); x = fmaf(p[4], w4, x); x = fmaf(p[5], w5, x);
        x = fmaf(p[6], w6, x);
        x = fmaxf(x, 0.0f);                       // relu
        const float xn = fmaf(x, scale, shift);   // batchnorm affine
        agg = fmaxf(agg, xn);                     // unmasked max (per reference)
        const float m = (float)mask[r];
        out[r * 32 + d] = xn * m;
    }
#pragma unroll 5
    for (int t = 0; t < TPTS; ++t) {
        const long r = (long)k * TPTS + t;
        out[r * 32 + 16 + d] = agg * (float)mask[r];
    }
}

extern "C" void kernel_launch(void* const* d_in, const int* in_sizes, int n_in,
                              void* d_out, int out_size, void* d_ws, size_t ws_size,
                              hipStream_t stream) {
    const float* in    = (const float*)d_in[0];
    const int*   mask  = (const int*)  d_in[1];
    const float* W     = (const float*)d_in[2];
    const float* bias  = (const float*)d_in[3];
    const float* gamma = (const float*)d_in[4];
    const float* beta  = (const float*)d_in[5];
    float* out   = (float*)d_out;
    float* stats = (float*)d_ws;   // 64 floats used

    vfe_zero_stats<<<1, 32, 0, stream>>>(stats);
    vfe_stats<<<512, 256, 0, stream>>>(in, W, bias, stats);          // 4096 waves
    vfe_finalize<<<1, 32, 0, stream>>>(gamma, beta, stats);
    vfe_output<<<(KVOX / 2 + 7) / 8, 256, 0, stream>>>(in, mask, W, bias, stats, out);
}